// DSTGCN_75642964017427
// MI455X (gfx1250) — compile-verified
//
#include <hip/hip_runtime.h>
#include <hip/hip_bf16.h>

// ---------------------------------------------------------------------------
// DSTGCN forward for MI455X (gfx1250, wave32).
//
//  * Dense linears: generic fp32 WMMA GEMM (V_WMMA_F32_16X16X4_F32).
//    Weight column-tile staged in LDS (zero-padded), shared by 8 waves/block;
//    A-row addresses clamped (garbage rows masked at the store), K-tail
//    handled with branchless selects -> tight loop: 2 global + 2 ds loads
//    per WMMA, no exec-mask churn.
//  * Graph aggregation (the roofline cost, ~2 GB through the 192 MB L2):
//    flat (edge,feature) scatter with GLOBAL_ATOMIC_ADD_F32; self-loop
//    contribution fused into the accumulator init.
//  * Conv1d(k=3,pad=1) on a length-1 sequence == linear with convW[:,:,1].
// ---------------------------------------------------------------------------

typedef float v2f __attribute__((ext_vector_type(2)));
typedef float v8f __attribute__((ext_vector_type(8)));

#define N_NODES 50000
#define N_EDGES 800000

// ---------------------------------------------------------------------------
// fp32 WMMA GEMM:  Y[M x C] = X[M x K] @ W[K x C]  (+bias)(+relu)
// grid = (ceil(ntm/8), ntc); 8 waves per block, all sharing one column tile.
// Dynamic LDS: Kp*16 floats holding W[:, tc*16 .. tc*16+16) zero-padded.
// ---------------------------------------------------------------------------
__global__ __launch_bounds__(256) void wmma_gemm_f32(
    const float* __restrict__ X, int ldx,
    const float* __restrict__ W,
    const float* __restrict__ bias,
    float* __restrict__ Y, int ldy,
    int M, int K, int C, int relu)
{
  extern __shared__ float smem[];               // [Kp][16]
  const int ntm = (M + 15) >> 4;
  const int tc  = blockIdx.y;
  const int Kp  = (K + 3) & ~3;

  // cooperative stage of the weight column tile (zero-padded)
  for (int i = threadIdx.x; i < Kp * 16; i += 256) {
    const int k = i >> 4;
    const int c = tc * 16 + (i & 15);
    smem[i] = (k < K && c < C) ? W[(size_t)k * C + c] : 0.f;
  }
  __syncthreads();

  const int tm = blockIdx.x * 8 + (threadIdx.x >> 5);
  if (tm >= ntm) return;                        // wave-uniform exit (after barrier)

  const int lane = threadIdx.x & 31;
  const int half = lane >> 4;
  const int l16  = lane & 15;
  const int row  = tm * 16 + l16;
  const int rowc = row < M ? row : M - 1;       // clamp: garbage rows masked at store
  const int col  = tc * 16 + l16;
  const float* __restrict__ Xr = X + (size_t)rowc * ldx;

  v8f acc = {0.f, 0.f, 0.f, 0.f, 0.f, 0.f, 0.f, 0.f};

  int k0 = 0;
  for (; k0 + 4 <= K; k0 += 4) {                // fully in-bounds: no predication
    v2f a, b;
    const int ka = k0 + half * 2;
    a.x = Xr[ka];
    a.y = Xr[ka + 1];
    b.x = smem[(k0 + half) * 16 + l16];
    b.y = smem[(k0 + 2 + half) * 16 + l16];
    acc = __builtin_amdgcn_wmma_f32_16x16x4_f32(
        false, a, false, b, (short)0, acc, false, false);
  }
  if (k0 < K) {                                 // K tail: clamped loads + selects
    const int ka  = k0 + half * 2;
    const int kc0 = ka     < K ? ka     : K - 1;
    const int kc1 = ka + 1 < K ? ka + 1 : K - 1;
    const float x0 = Xr[kc0];
    const float x1 = Xr[kc1];
    v2f a, b;
    a.x = (ka     < K) ? x0 : 0.f;
    a.y = (ka + 1 < K) ? x1 : 0.f;
    b.x = smem[(k0 + half) * 16 + l16];         // LDS already zero-padded
    b.y = smem[(k0 + 2 + half) * 16 + l16];
    acc = __builtin_amdgcn_wmma_f32_16x16x4_f32(
        false, a, false, b, (short)0, acc, false, false);
  }

#pragma unroll
  for (int j = 0; j < 8; ++j) {
    const int m = tm * 16 + j + half * 8;
    if (m < M && col < C) {
      float v = acc[j];
      if (bias) v += bias[col];
      if (relu) v = fmaxf(v, 0.f);
      Y[(size_t)m * ldy + col] = v;
    }
  }
}

// ---------------------------------------------------------------------------
// Elementwise / graph kernels
// ---------------------------------------------------------------------------
__global__ void fill_f32(float* p, float v, int n) {
  int i = blockIdx.x * blockDim.x + threadIdx.x;
  if (i < n) p[i] = v;
}

__global__ void deg_accum(const int* __restrict__ dst, float* deg, int E) {
  int e = blockIdx.x * blockDim.x + threadIdx.x;
  if (e < E) atomicAdd(&deg[dst[e]], 1.f);
}

__global__ void deg_norm(const float* __restrict__ deg,
                         float* __restrict__ dis, float* __restrict__ dinv, int n) {
  int i = blockIdx.x * blockDim.x + threadIdx.x;
  if (i < n) { float d = deg[i]; dis[i] = rsqrtf(d); dinv[i] = 1.f / d; }
}

__global__ void edge_norm(const int* __restrict__ src, const int* __restrict__ dst,
                          const float* __restrict__ dis, float* __restrict__ enrm, int E) {
  int e = blockIdx.x * blockDim.x + threadIdx.x;
  if (e < E) enrm[e] = dis[src[e]] * dis[dst[e]];
}

// agg[n,f] = y[n,f] * dinv[n]   (self-loop contribution seeds the accumulator)
__global__ void selfloop_init(float* __restrict__ agg, const float* __restrict__ y,
                              const float* __restrict__ dinv, int N, int F) {
  int idx = blockIdx.x * blockDim.x + threadIdx.x;
  if (idx >= N * F) return;
  agg[idx] = y[idx] * dinv[idx / F];
}

// agg[dst, f] += y[src, f] * enrm[e], flattened over (edge, feature)
__global__ void gcn_scatter(const float* __restrict__ y,
                            const int* __restrict__ src, const int* __restrict__ dst,
                            const float* __restrict__ enrm,
                            float* __restrict__ agg, int E, int F) {
  int idx = blockIdx.x * blockDim.x + threadIdx.x;
  if (idx >= E * F) return;
  int e = idx / F;
  int f = idx - e * F;
  float v = y[(size_t)src[e] * F + f] * enrm[e];
  atomicAdd(&agg[(size_t)dst[e] * F + f], v);
}

// out = relu(agg + bias[f])
__global__ void bias_relu(const float* __restrict__ agg, const float* __restrict__ bias,
                          float* __restrict__ out, int N, int F) {
  int idx = blockIdx.x * blockDim.x + threadIdx.x;
  if (idx >= N * F) return;
  out[idx] = fmaxf(agg[idx] + bias[idx % F], 0.f);
}

__global__ void add_inplace(float* __restrict__ a, const float* __restrict__ b, int n) {
  int i = blockIdx.x * blockDim.x + threadIdx.x;
  if (i < n) a[i] += b[i];
}

__global__ void copy_strided(float* __restrict__ dst, int ldd,
                             const float* __restrict__ src, int lds, int N, int F) {
  int idx = blockIdx.x * blockDim.x + threadIdx.x;
  if (idx >= N * F) return;
  int n = idx / F;
  int f = idx - n * F;
  dst[(size_t)n * ldd + f] = src[(size_t)n * lds + f];
}

// wt[i*f + o] = convW[o, i, 1]   (middle tap of k=3 conv on length-1 seq)
__global__ void extract_conv_mid(const float* __restrict__ cw, float* __restrict__ wt, int f) {
  int i = threadIdx.x;
  if (i < f * f) {
    int o  = i % f;
    int in = i / f;
    wt[in * f + o] = cw[o * f * 3 + in * 3 + 1];
  }
}

// feat[n,0:20]=s_out, feat[n,20:22]=avgpool12(hbuf[n,0:24]), feat[n,22:42]=e_out; relu
__global__ void concat_relu(const float* __restrict__ s20, const float* __restrict__ hbuf,
                            const float* __restrict__ e20, float* __restrict__ feat, int N) {
  int idx = blockIdx.x * blockDim.x + threadIdx.x;
  if (idx >= N * 42) return;
  int n = idx / 42;
  int j = idx - n * 42;
  float v;
  if (j < 20) {
    v = s20[(size_t)n * 20 + j];
  } else if (j < 22) {
    const float* h = hbuf + (size_t)n * 32 + (j - 20) * 12;
    float s = 0.f;
#pragma unroll
    for (int k = 0; k < 12; ++k) s += h[k];
    v = s * (1.f / 12.f);
  } else {
    v = e20[(size_t)n * 20 + (j - 22)];
  }
  feat[idx] = fmaxf(v, 0.f);
}

// ---------------------------------------------------------------------------
// Host orchestration
// ---------------------------------------------------------------------------
extern "C" void kernel_launch(void* const* d_in, const int* in_sizes, int n_in,
                              void* d_out, int out_size, void* d_ws, size_t ws_size,
                              hipStream_t stream) {
  const int N = N_NODES;
  const int E = N_EDGES;

  const float* spat = (const float*)d_in[0];   // [N,103]
  const float* temp = (const float*)d_in[1];   // [N,1,2]
  const float* ext  = (const float*)d_in[2];   // [N,22]
  const int*   eidx = (const int*)d_in[3];     // [2,E]
  const int* src = eidx;
  const int* dst = eidx + E;

  // param leaves in jax pytree order: ee, out, se, sgcn, st (st dicts: convW, convb, gcn)
  auto P = [&](int i) -> const float* { return (const float*)d_in[i]; };
  const int EE_W0 = 4, EE_B0 = 5, EE_W1 = 6, EE_B1 = 7, EE_W2 = 8, EE_B2 = 9;
  const int OUT_W = 10, OUT_B = 11;
  const int SE_W0 = 12, SE_B0 = 13, SE_W1 = 14, SE_B1 = 15;
  const int SG0 = 16;   // 4 layers (W,b): 16..23
  const int SG1 = 24;   // 4 layers: 24..31
  const int SG2 = 32;   // 5 layers: 32..41
  const int ST0 = 42;   // 4 blocks x (convW, convb, 3x(W,b)) = 8 leaves each

  // workspace partition (floats)
  float* ws = (float*)d_ws;
  size_t o = 0;
  float* deg    = ws + o; o += N;
  float* dis    = ws + o; o += N;
  float* dinv   = ws + o; o += N;
  float* enrm   = ws + o; o += E;
  float* bufY   = ws + o; o += (size_t)N * 32;  // gcn linear output
  float* bufAgg = ws + o; o += (size_t)N * 32;  // scatter accumulator
  float* bufX1  = ws + o; o += (size_t)N * 32;  // ping
  float* bufX2  = ws + o; o += (size_t)N * 32;  // pong
  float* sres   = ws + o; o += (size_t)N * 32;  // residual s
  float* sout   = ws + o; o += (size_t)N * 20;  // spatial branch output
  float* tbY    = ws + o; o += (size_t)N * 16;  // temporal gcn linear out
  float* tbX1   = ws + o; o += (size_t)N * 16;
  float* tbX2   = ws + o; o += (size_t)N * 16;
  float* hbuf   = ws + o; o += (size_t)N * 32;  // temporal channel concat
  float* bufFC  = ws + o; o += (size_t)N * 40;  // FC hidden
  float* bufFC2 = ws + o; o += (size_t)N * 20;
  float* eout   = ws + o; o += (size_t)N * 20;
  float* feat   = ws + o; o += (size_t)N * 42;
  float* wtmp   = ws + o; o += 256;             // conv mid-tap weights
  if (o * sizeof(float) > ws_size) return;      // leave d_out poisoned -> visible

  auto grid1 = [](int n) { return dim3((unsigned)((n + 255) / 256)); };
  auto gemm = [&](const float* X, int ldx, const float* W, const float* bias,
                  float* Y, int ldy, int M, int K, int C, int relu) {
    const int ntm = (M + 15) / 16;
    const int ntc = (C + 15) / 16;
    const int Kp  = (K + 3) & ~3;
    wmma_gemm_f32<<<dim3((unsigned)((ntm + 7) / 8), (unsigned)ntc), dim3(256),
                    (size_t)Kp * 16 * sizeof(float), stream>>>(
        X, ldx, W, bias, Y, ldy, M, K, C, relu);
  };
  auto gcn_layer = [&](const float* x, int ldx, int fin,
                       const float* W, const float* b,
                       float* ybuf, float* out, int fout) {
    gemm(x, ldx, W, nullptr, ybuf, fout, N, fin, fout, 0);
    selfloop_init<<<grid1(N * fout), 256, 0, stream>>>(bufAgg, ybuf, dinv, N, fout);
    gcn_scatter<<<grid1(E * fout), 256, 0, stream>>>(ybuf, src, dst, enrm, bufAgg, E, fout);
    bias_relu<<<grid1(N * fout), 256, 0, stream>>>(bufAgg, b, out, N, fout);
  };

  // ---- degree / symmetric normalization ----
  fill_f32<<<grid1(N), 256, 0, stream>>>(deg, 1.f, N);           // self loop
  deg_accum<<<grid1(E), 256, 0, stream>>>(dst, deg, E);
  deg_norm<<<grid1(N), 256, 0, stream>>>(deg, dis, dinv, N);
  edge_norm<<<grid1(E), 256, 0, stream>>>(src, dst, dis, enrm, E);

  // ---- spatial branch ----
  gemm(spat, 103, P(SE_W0), P(SE_B0), bufFC, 40, N, 103, 40, 1);
  gemm(bufFC, 40, P(SE_W1), P(SE_B1), sres, 30, N, 40, 30, 0);   // s [N,30]

  for (int blk = 0; blk < 2; ++blk) {                            // residual GCN blocks
    int base = (blk == 0) ? SG0 : SG1;
    const float* x = sres; int ldx = 30; int pp = 0;
    for (int i = 0; i < 4; ++i) {
      float* out = pp ? bufX2 : bufX1;
      gcn_layer(x, ldx, 30, P(base + 2 * i), P(base + 2 * i + 1), bufY, out, 30);
      x = out; ldx = 30; pp ^= 1;
    }
    add_inplace<<<grid1(N * 30), 256, 0, stream>>>(sres, x, N * 30);
  }
  {                                                              // plain chain 30->...->20
    const int dims[6] = {30, 28, 26, 14, 22, 20};
    const float* x = sres; int ldx = 30; int pp = 0;
    for (int i = 0; i < 5; ++i) {
      float* out = (i == 4) ? sout : (pp ? bufX2 : bufX1);
      gcn_layer(x, ldx, dims[i], P(SG2 + 2 * i), P(SG2 + 2 * i + 1), bufY, out, dims[i + 1]);
      x = out; ldx = dims[i + 1]; pp ^= 1;
    }
  }

  // ---- temporal branch (channel-concat growth 2->4->8->16->32) ----
  copy_strided<<<grid1(N * 2), 256, 0, stream>>>(hbuf, 32, temp, 2, N, 2);
  int cin = 2;
  for (int b = 0; b < 4; ++b) {
    int base = ST0 + b * 8;                  // convW, convb, 3x(W,b)
    const float* x = hbuf; int ldx = 32; int pp = 0;
    for (int j = 0; j < 3; ++j) {            // gcn stack: cin->cin x3, relu each
      float* out = pp ? tbX2 : tbX1;
      gcn_layer(x, ldx, cin, P(base + 2 + 2 * j), P(base + 3 + 2 * j), tbY, out, cin);
      x = out; ldx = cin; pp ^= 1;
    }
    // conv1d(k=3,pad=1) on len-1 == linear with middle tap; write into hbuf cols [cin,2cin)
    extract_conv_mid<<<dim3(1), dim3(256), 0, stream>>>(P(base), wtmp, cin);
    gemm(x, ldx, wtmp, P(base + 1), hbuf + cin, 32, N, cin, cin, 0);
    cin *= 2;
  }

  // ---- external branch ----
  gemm(ext, 22, P(EE_W0), P(EE_B0), bufFC, 19, N, 22, 19, 1);
  gemm(bufFC, 19, P(EE_W1), P(EE_B1), bufFC2, 10, N, 19, 10, 1);
  gemm(bufFC2, 10, P(EE_W2), P(EE_B2), eout, 20, N, 10, 20, 0);

  // ---- concat + relu + avgpool12, then final linear 42->2 ----
  concat_relu<<<grid1(N * 42), 256, 0, stream>>>(sout, hbuf, eout, feat, N);
  gemm(feat, 42, P(OUT_W), P(OUT_B), (float*)d_out, 2, N, 42, 2, 0);
}